// SynapticDepression_38869454029216
// MI455X (gfx1250) — compile-verified
//
#include <hip/hip_runtime.h>
#include <hip/hip_bf16.h>
#include <math.h>

// ---------------------------------------------------------------------------
// SynapticDepression for MI455X (gfx1250, wave32).
// Pass 1: block-level partial reductions of (I.I, P.P, I.P)   [bandwidth bound]
// Pass 2: 1 wave: finalize partials -> dishabituation flag    [uses v_wmma]
// Pass 3: elementwise update, b128 loads, nontemporal stores  [bandwidth bound]
// ---------------------------------------------------------------------------

#define NB_RED 2048
#define TPB    256
#define NB_APPLY 8192

typedef float v2f __attribute__((ext_vector_type(2)));
typedef float v4f __attribute__((ext_vector_type(4)));
typedef float v8f __attribute__((ext_vector_type(8)));

__device__ __forceinline__ float wave_reduce_add(float v) {
#pragma unroll
    for (int m = 16; m >= 1; m >>= 1) v += __shfl_xor(v, m, 32);
    return v;
}

// ---------------- Pass 1: partial reductions --------------------------------
__global__ void syn_reduce_kernel(const float* __restrict__ I,
                                  const float* __restrict__ P,
                                  float* __restrict__ partials, int n) {
    const int n4     = n >> 2;
    const v4f* I4    = (const v4f*)I;
    const v4f* P4    = (const v4f*)P;
    const int tid    = blockIdx.x * blockDim.x + threadIdx.x;
    const int stride = gridDim.x * blockDim.x;

    float sii = 0.f, spp = 0.f, sip = 0.f;
    for (int i = tid; i < n4; i += stride) {
        v4f a = I4[i];
        v4f b = P4[i];
#pragma unroll
        for (int c = 0; c < 4; ++c) {
            sii = fmaf(a[c], a[c], sii);
            spp = fmaf(b[c], b[c], spp);
            sip = fmaf(a[c], b[c], sip);
        }
    }
    // scalar tail (n not multiple of 4)
    for (int i = (n4 << 2) + tid; i < n; i += stride) {
        float a = I[i], b = P[i];
        sii = fmaf(a, a, sii);
        spp = fmaf(b, b, spp);
        sip = fmaf(a, b, sip);
    }

    // wave32 shuffle reduction, then LDS across the 8 waves of this block
    sii = wave_reduce_add(sii);
    spp = wave_reduce_add(spp);
    sip = wave_reduce_add(sip);

    __shared__ float sm[3][TPB / 32];
    const int lane = threadIdx.x & 31;
    const int wv   = threadIdx.x >> 5;
    if (lane == 0) { sm[0][wv] = sii; sm[1][wv] = spp; sm[2][wv] = sip; }
    __syncthreads();

    if (wv == 0) {
        float a = (lane < TPB / 32) ? sm[0][lane] : 0.f;
        float b = (lane < TPB / 32) ? sm[1][lane] : 0.f;
        float c = (lane < TPB / 32) ? sm[2][lane] : 0.f;
#pragma unroll
        for (int m = 4; m >= 1; m >>= 1) {
            a += __shfl_xor(a, m, 32);
            b += __shfl_xor(b, m, 32);
            c += __shfl_xor(c, m, 32);
        }
        if (lane == 0) {
            partials[0 * gridDim.x + blockIdx.x] = a;
            partials[1 * gridDim.x + blockIdx.x] = b;
            partials[2 * gridDim.x + blockIdx.x] = c;
        }
    }
}

// ---------------- Pass 2: finalize (single wave, WMMA-assisted) -------------
// Layout-robust WMMA sum: A == 1.0 in every slot  =>  D[m][n] = sum_k B[k][n].
// C-VGPR0 across all 32 lanes covers two full D rows, each of which sums to
// the total of all 64 B slots, so (32-lane sum of c[0]) * 0.5 == sum(B).
__global__ void syn_finalize_kernel(const float* __restrict__ partials, int nb,
                                    float* __restrict__ scalars) {
    const int lane = threadIdx.x;  // exactly 32 threads: EXEC is all ones
    float acc[3];
#pragma unroll
    for (int q = 0; q < 3; ++q) {
        float s = 0.f;
        for (int i = lane; i < nb; i += 32) s += partials[q * nb + i];
        acc[q] = s;
    }

    v2f ones; ones[0] = 1.0f; ones[1] = 1.0f;
    float tot[3];
#pragma unroll
    for (int q = 0; q < 3; ++q) {
        v2f b; b[0] = acc[q]; b[1] = 0.0f;
        v8f c = {};
        // D = A(ones) x B(values) + 0   -> every D row sums to sum(acc over lanes)
        c = __builtin_amdgcn_wmma_f32_16x16x4_f32(
                /*neg_a=*/false, ones, /*neg_b=*/false, b,
                /*c_mod=*/(short)0, c, /*reuse_a=*/false, /*reuse_b=*/false);
        float v = c[0];
        v = wave_reduce_add(v);   // two full rows -> 2x total
        tot[q] = v * 0.5f;
    }

    if (lane == 0) {
        float inp_norm  = sqrtf(tot[0]) + 1e-8f;
        float prev_norm = sqrtf(tot[1]) + 1e-8f;
        float cosine    = tot[2] / (inp_norm * prev_norm);
        bool apply = (inp_norm > 0.5f) && (prev_norm > 0.5f) && (cosine < 0.7f);
        scalars[0] = apply ? 1.0f : 0.0f;
    }
}

// ---------------- Pass 3: elementwise update --------------------------------
__device__ __forceinline__ float syn_apply_one(float I, float dep, float P,
                                               bool flag) {
    if (flag && fabsf(I - P) > 0.5f) dep = fminf(dep + 0.3f, 1.0f);
    float active = (fabsf(I) > 0.1f) ? 1.0f : 0.0f;
    dep = dep * (1.0f - 0.02f * active);
    dep = dep + 0.01f * (1.0f - dep);
    dep = fminf(fmaxf(dep, 0.3f), 1.0f);
    return I * dep;
}

__global__ void syn_apply_kernel(const float* __restrict__ I,
                                 const float* __restrict__ D,
                                 const float* __restrict__ P,
                                 const float* __restrict__ scalars,
                                 float* __restrict__ out, int n) {
    const bool flag  = (scalars[0] != 0.0f);  // wave-uniform scalar load
    const int n4     = n >> 2;
    const v4f* I4    = (const v4f*)I;
    const v4f* D4    = (const v4f*)D;
    const v4f* P4    = (const v4f*)P;
    v4f* O4          = (v4f*)out;
    const int tid    = blockIdx.x * blockDim.x + threadIdx.x;
    const int stride = gridDim.x * blockDim.x;

    for (int i = tid; i < n4; i += stride) {
        v4f a = I4[i];
        v4f d = D4[i];
        v4f p = P4[i];
        v4f r;
#pragma unroll
        for (int c = 0; c < 4; ++c) r[c] = syn_apply_one(a[c], d[c], p[c], flag);
        __builtin_nontemporal_store(r, &O4[i]);  // streamed once, skip caches
    }
    for (int i = (n4 << 2) + tid; i < n; i += stride) {
        float r = syn_apply_one(I[i], D[i], P[i], flag);
        __builtin_nontemporal_store(r, &out[i]);
    }
}

// ---------------------------------------------------------------------------
extern "C" void kernel_launch(void* const* d_in, const int* in_sizes, int n_in,
                              void* d_out, int out_size, void* d_ws, size_t ws_size,
                              hipStream_t stream) {
    const float* I  = (const float*)d_in[0];   // I_ext
    const float* Dp = (const float*)d_in[1];   // depletion
    const float* P  = (const float*)d_in[2];   // prev_input
    float* out      = (float*)d_out;
    const int n     = in_sizes[0];

    float* wsf      = (float*)d_ws;
    float* partials = wsf;                 // 3 * NB_RED floats
    float* scalars  = wsf + 3 * NB_RED;    // 1 float (dishabituation flag)

    syn_reduce_kernel  <<<NB_RED,   TPB, 0, stream>>>(I, P, partials, n);
    syn_finalize_kernel<<<1,        32,  0, stream>>>(partials, NB_RED, scalars);
    syn_apply_kernel   <<<NB_APPLY, TPB, 0, stream>>>(I, Dp, P, scalars, out, n);
}